// MyScore_22754736735003
// MI455X (gfx1250) — compile-verified
//
#include <hip/hip_runtime.h>
#include <hip/hip_bf16.h>
#include <math.h>

typedef __attribute__((ext_vector_type(2))) float v2f;
typedef __attribute__((ext_vector_type(8))) float v8f;

#define IN_CH 256

__device__ __forceinline__ float sigmoidf_(float v) {
    return 1.0f / (1.0f + __expf(-v));
}

// Zero the accumulators we atomically update (deg histogram + gcn sum).
__global__ void init_kernel(unsigned int* __restrict__ deg,
                            float* __restrict__ gcn, int n) {
    int i = blockIdx.x * blockDim.x + threadIdx.x;
    if (i < n) { deg[i] = 0u; gcn[i] = 0.0f; }
}

// In-degree histogram over edge targets (int64 indices, values < 2^31).
__global__ void hist_kernel(const long long* __restrict__ dst,
                            unsigned int* __restrict__ deg, int e) {
    int i = blockIdx.x * blockDim.x + threadIdx.x;
    if (i < e) atomicAdd(&deg[(int)dst[i]], 1u);
}

// dinv = rsqrt(in_deg + 1)  (GCN degree includes the self loop)
__global__ void dinv_kernel(const unsigned int* __restrict__ deg,
                            float* __restrict__ dinv, int n) {
    int i = blockIdx.x * blockDim.x + threadIdx.x;
    if (i < n) dinv[i] = rsqrtf((float)deg[i] + 1.0f);
}

// Fused tri-matvec via V_WMMA_F32_16X16X4_F32.
// A (16x4, M x K): M rows 0..2 = W_mlp / W_lin / W_gcn, rows 3..15 zero.
//   lane<16:  a = {A[l][k],   A[l][k+1]}     lane>=16: a = {A[l-16][k+2], A[l-16][k+3]}
// B (4x16, K x N): N = node within the 16-node tile (mirrors A with M<->N).
//   lane<16:  b = {x[node_l][k],   x[node_l][k+1]}
//   lane>=16: b = {x[node_l][k+2], x[node_l][k+3]}
// D (16x16 f32): VGPR r, lanes 0..15 -> D[M=r][N=lane]; we keep M=0..2 on lanes 0..15.
__global__ void matvec_wmma_kernel(const float* __restrict__ x,
                                   const float* __restrict__ Wmlp,
                                   const float* __restrict__ Wlin,
                                   const float* __restrict__ Wgcn,
                                   float* __restrict__ xw_mlp,
                                   float* __restrict__ xw_lin,
                                   float* __restrict__ xw_gcn,
                                   int n) {
    const int wave = threadIdx.x >> 5;          // wave32
    const int lane = threadIdx.x & 31;
    const int tile = blockIdx.x * (blockDim.x >> 5) + wave;
    if (tile * 16 >= n) return;                 // wave-uniform: EXEC all-1s below
    const int l16  = lane & 15;
    const int half = lane >> 4;                 // selects K pair {0,1} vs {2,3}
    const int node = tile * 16 + l16;

    const float* xrow  = x + (size_t)node * IN_CH;
    const float* wrow  = (l16 == 1) ? Wlin : (l16 == 2) ? Wgcn : Wmlp;
    const float  amask = (l16 < 3) ? 1.0f : 0.0f;  // zero rows 3..15, no divergence

    v8f c = {};
#pragma unroll 8
    for (int k = 0; k < IN_CH; k += 4) {
        v2f a = *(const v2f*)(wrow + k + 2 * half);
        a.x *= amask;
        a.y *= amask;
        v2f b = *(const v2f*)(xrow + k + 2 * half);
        // D = A x B + C ; 8 args: (neg_a, A, neg_b, B, c_mod, C, reuse_a, reuse_b)
        c = __builtin_amdgcn_wmma_f32_16x16x4_f32(
                false, a, false, b, (short)0, c, false, false);
    }
    if (lane < 16) {
        xw_mlp[node] = c[0];   // M=0
        xw_lin[node] = c[1];   // M=1
        xw_gcn[node] = c[2];   // M=2
    }
}

// GCN message scatter: gcn[dst] += dinv[src]*dinv[dst]*xw_gcn[src]
__global__ void edge_kernel(const long long* __restrict__ ei,
                            const float* __restrict__ dinv,
                            const float* __restrict__ xw_gcn,
                            float* __restrict__ gcn, int e) {
    int i = blockIdx.x * blockDim.x + threadIdx.x;
    if (i < e) {
        int s = (int)ei[i];
        int d = (int)ei[(size_t)e + i];
        atomicAdd(&gcn[d], dinv[s] * dinv[d] * xw_gcn[s]);
    }
}

// Per-node: sigmoids, self-loop term, 3x3 attention, softmax, weighted sum.
__global__ void final_kernel(const unsigned int* __restrict__ deg,
                             const float* __restrict__ dinv,
                             const float* __restrict__ xw_mlp,
                             const float* __restrict__ xw_lin,
                             const float* __restrict__ xw_gcn,
                             const float* __restrict__ gcn,
                             const float* __restrict__ alpha_p,
                             const float* __restrict__ beta_p,
                             const float* __restrict__ W_att,
                             const float* __restrict__ b_att,
                             float* __restrict__ out, int n) {
    int i = blockIdx.x * blockDim.x + threadIdx.x;
    if (i >= n) return;
    const float alpha = alpha_p[0];
    const float beta  = beta_p[0];

    const float s1 = sigmoidf_(alpha * sqrtf((float)deg[i]) + beta);
    const float s2 = sigmoidf_(xw_mlp[i]);
    const float di = dinv[i];
    const float g  = gcn[i] + di * di * xw_gcn[i];   // add self-loop message
    const float s3 = sigmoidf_(g + xw_lin[i]);

    float t0 = b_att[0] + W_att[0] * s1 + W_att[1] * s2 + W_att[2] * s3;
    float t1 = b_att[1] + W_att[3] * s1 + W_att[4] * s2 + W_att[5] * s3;
    float t2 = b_att[2] + W_att[6] * s1 + W_att[7] * s2 + W_att[8] * s3;
    float m  = fmaxf(t0, fmaxf(t1, t2));
    float e0 = __expf(t0 - m), e1 = __expf(t1 - m), e2 = __expf(t2 - m);
    float inv = 1.0f / (e0 + e1 + e2);
    out[i] = (e0 * s1 + e1 * s2 + e2 * s3) * inv;
}

extern "C" void kernel_launch(void* const* d_in, const int* in_sizes, int n_in,
                              void* d_out, int out_size, void* d_ws, size_t ws_size,
                              hipStream_t stream) {
    const float*     x    = (const float*)d_in[0];
    const long long* ei   = (const long long*)d_in[1];   // int64 [2, E]
    const float*     a_p  = (const float*)d_in[2];
    const float*     b_p  = (const float*)d_in[3];
    const float*     Wmlp = (const float*)d_in[4];
    const float*     Wlin = (const float*)d_in[5];
    const float*     Wgcn = (const float*)d_in[6];
    const float*     Watt = (const float*)d_in[7];
    const float*     batt = (const float*)d_in[8];
    float*           out  = (float*)d_out;

    const int n = in_sizes[0] / IN_CH;   // 100000
    const int e = in_sizes[1] / 2;       // 3200000

    float* ws = (float*)d_ws;
    unsigned int* deg = (unsigned int*)ws;        // [n]
    float* dinv   = ws + (size_t)n;               // [n]
    float* xw_mlp = ws + (size_t)2 * n;           // [n]
    float* xw_lin = ws + (size_t)3 * n;           // [n]
    float* xw_gcn = ws + (size_t)4 * n;           // [n]
    float* gcn    = ws + (size_t)5 * n;           // [n]

    const int TB = 256;
    init_kernel<<<(n + TB - 1) / TB, TB, 0, stream>>>(deg, gcn, n);
    hist_kernel<<<(e + TB - 1) / TB, TB, 0, stream>>>(ei + (size_t)e, deg, e);

    const int tiles = (n + 15) / 16;              // 6250, all full (n % 16 == 0)
    const int wavesPerBlock = TB / 32;            // 8 waves -> 8 tiles per block
    matvec_wmma_kernel<<<(tiles + wavesPerBlock - 1) / wavesPerBlock, TB, 0, stream>>>(
        x, Wmlp, Wlin, Wgcn, xw_mlp, xw_lin, xw_gcn, n);

    dinv_kernel<<<(n + TB - 1) / TB, TB, 0, stream>>>(deg, dinv, n);
    edge_kernel<<<(e + TB - 1) / TB, TB, 0, stream>>>(ei, dinv, xw_gcn, gcn, e);
    final_kernel<<<(n + TB - 1) / TB, TB, 0, stream>>>(
        deg, dinv, xw_mlp, xw_lin, xw_gcn, gcn, a_p, b_p, Watt, batt, out, n);
}